// GAT_52235392254450
// MI455X (gfx1250) — compile-verified
//
#include <hip/hip_runtime.h>
#include <hip/hip_bf16.h>
#include <math.h>

#define NEG_SLOPE 0.2f
#define GAT_EPS 1e-16f

typedef __attribute__((ext_vector_type(2))) float v2f;
typedef __attribute__((ext_vector_type(8))) float v8f;

// ---- orderable-uint encoding for float atomicMax (handles negatives) ----
__device__ __forceinline__ unsigned f2ord(float f) {
  unsigned u = __float_as_uint(f);
  return (u & 0x80000000u) ? ~u : (u | 0x80000000u);
}
__device__ __forceinline__ float ord2f(unsigned u) {
  return __uint_as_float((u & 0x80000000u) ? (u & 0x7fffffffu) : ~u);
}

// =====================================================================
// K1: xp = h @ W   (N x 64) = (N x 64)(64 x 64), fp32 WMMA 16x16x4.
// Block = 256 threads = 8 waves; block covers 32 rows x 64 cols.
// No per-lane predication: out-of-range rows are CLAMPED (their results
// are discarded at store time), so EXEC stays all-1s and loads are
// branch-free. All loads hoisted ahead of the 16 back-to-back WMMAs.
// A layout (ISA 32-bit A 16x4): lane l<16 holds K=(4kk+0,4kk+1) of row l,
// lane l>=16 holds K=(4kk+2,4kk+3) of row l-16. B mirrored (row-striped).
// =====================================================================
__global__ void gat_gemm_wmma(const float* __restrict__ h,
                              const float* __restrict__ W,
                              float* __restrict__ xp, int N) {
  const int wave = threadIdx.x >> 5;
  const int lane = threadIdx.x & 31;
  const int half = lane >> 4;
  const int lm   = lane & 15;
  const int base_m = (blockIdx.x * 2 + (wave >> 2)) * 16;
  const int base_n = (wave & 3) * 16;
  if (base_m >= N) return;  // wave-uniform exit; surviving waves keep EXEC all-1s

  int rowA = base_m + lm;
  if (rowA >= N) rowA = N - 1;              // clamp: never stored, always valid
  const float* __restrict__ hrow = h + (size_t)rowA * 64 + half * 2;
  const float* __restrict__ wcol = W + base_n + lm + (size_t)half * 2 * 64;

  // ---- hoist all global loads (A: 16x b64, B: 32x b32) ----
  v2f a[16];
#pragma unroll
  for (int kk = 0; kk < 16; ++kk)
    a[kk] = *(const v2f*)(hrow + kk * 4);

  float bx[16], by[16];
#pragma unroll
  for (int kk = 0; kk < 16; ++kk) {
    bx[kk] = wcol[(size_t)kk * 4 * 64];
    by[kk] = wcol[(size_t)(kk * 4 + 1) * 64];
  }

  // ---- 16 back-to-back fp32 WMMAs ----
  v8f acc = {};
#pragma unroll
  for (int kk = 0; kk < 16; ++kk) {
    v2f b;
    b.x = bx[kk];
    b.y = by[kk];
    acc = __builtin_amdgcn_wmma_f32_16x16x4_f32(
        /*neg_a=*/false, a[kk], /*neg_b=*/false, b,
        /*c_mod=*/(short)0, acc, /*reuse_a=*/false, /*reuse_b=*/false);
  }

#pragma unroll
  for (int r = 0; r < 8; ++r) {
    const int m = base_m + r + half * 8;
    if (m < N) xp[(size_t)m * 64 + base_n + lm] = acc[r];
  }
}

// =====================================================================
// K2: attention logits per (node, head): al = <xp[n,h,:], a[h,:]>
// =====================================================================
__global__ void gat_logits(const float* __restrict__ xp,
                           const float* __restrict__ a_src,
                           const float* __restrict__ a_dst,
                           float* __restrict__ als, float* __restrict__ ald,
                           int N, int H, int C) {
  const int t = blockIdx.x * blockDim.x + threadIdx.x;
  if (t >= N * H) return;
  const int n = t / H, hd = t - n * H;
  const float* v  = xp + (size_t)n * 64 + hd * C;
  const float* as = a_src + hd * C;
  const float* ad = a_dst + hd * C;
  float s = 0.f, d = 0.f;
  for (int c = 0; c < C; ++c) { s = fmaf(v[c], as[c], s); d = fmaf(v[c], ad[c], d); }
  als[t] = s;
  ald[t] = d;
}

// =====================================================================
// K3: init segment-max (encoded 0 == -NaN sentinel, below every real key),
//     denom = 0, accumulator = 0
// =====================================================================
__global__ void gat_init(unsigned* __restrict__ menc, float* __restrict__ denom,
                         float* __restrict__ acc, int N, int H) {
  const int t = blockIdx.x * blockDim.x + threadIdx.x;
  if (t < N * H) { menc[t] = 0u; denom[t] = 0.0f; }
  if (t < N * 64) acc[t] = 0.0f;
}

// =====================================================================
// K4: per-edge-per-head segment max of leaky_relu(al_s[src]+al_d[dst])
//     edge ids >= E are self loops (src = dst = e - E)
// =====================================================================
__global__ void gat_edge_max(const int* __restrict__ ei, int E, int N, int H,
                             const float* __restrict__ als,
                             const float* __restrict__ ald,
                             unsigned* __restrict__ menc) {
  const int t = blockIdx.x * blockDim.x + threadIdx.x;
  const int tot = (E + N) * H;
  if (t >= tot) return;
  const int e = t / H, hd = t - e * H;
  int s, d;
  if (e < E) { s = ei[e]; d = ei[E + e]; } else { s = d = e - E; }
  float v = als[s * H + hd] + ald[d * H + hd];
  v = v > 0.0f ? v : NEG_SLOPE * v;
  atomicMax(&menc[d * H + hd], f2ord(v));
}

// =====================================================================
// K5: per-edge-per-head denom += exp(val - max)  (recompute val; cheaper
//     than storing E*H floats)
// =====================================================================
__global__ void gat_edge_sum(const int* __restrict__ ei, int E, int N, int H,
                             const float* __restrict__ als,
                             const float* __restrict__ ald,
                             const unsigned* __restrict__ menc,
                             float* __restrict__ denom) {
  const int t = blockIdx.x * blockDim.x + threadIdx.x;
  const int tot = (E + N) * H;
  if (t >= tot) return;
  const int e = t / H, hd = t - e * H;
  int s, d;
  if (e < E) { s = ei[e]; d = ei[E + e]; } else { s = d = e - E; }
  float v = als[s * H + hd] + ald[d * H + hd];
  v = v > 0.0f ? v : NEG_SLOPE * v;
  const float ev = __expf(v - ord2f(menc[d * H + hd]));
  unsafeAtomicAdd(&denom[d * H + hd], ev);
}

// =====================================================================
// K6: message scatter. 16 lanes per edge; each lane owns 4 consecutive
//     channels (one head for C in {8,64}): float4 gather of xp[src],
//     4x global_atomic_add_f32 into acc[dst].
// =====================================================================
__global__ void gat_edge_msg(const int* __restrict__ ei, int E, int N,
                             int H, int C,
                             const float* __restrict__ xp,
                             const float* __restrict__ als,
                             const float* __restrict__ ald,
                             const unsigned* __restrict__ menc,
                             const float* __restrict__ denom,
                             float* __restrict__ acc) {
  const int t = blockIdx.x * blockDim.x + threadIdx.x;
  const int tot = (E + N) * 16;
  if (t >= tot) return;
  const int e = t >> 4, l = t & 15;
  int s, d;
  if (e < E) { s = ei[e]; d = ei[E + e]; } else { s = d = e - E; }
  const int ch0 = l * 4;
  const int hd  = ch0 / C;
  float v = als[s * H + hd] + ald[d * H + hd];
  v = v > 0.0f ? v : NEG_SLOPE * v;
  const float ev    = __expf(v - ord2f(menc[d * H + hd]));
  const float alpha = ev / (denom[d * H + hd] + GAT_EPS);
  const float4 xs = *(const float4*)(xp + (size_t)s * 64 + ch0);
  float* o = acc + (size_t)d * 64 + ch0;
  unsafeAtomicAdd(o + 0, xs.x * alpha);
  unsafeAtomicAdd(o + 1, xs.y * alpha);
  unsafeAtomicAdd(o + 2, xs.z * alpha);
  unsafeAtomicAdd(o + 3, xs.w * alpha);
}

// =====================================================================
// K7a: h_next = elu(acc + bias)      (layers 1,2)
// K7b: out    = acc + bias + x_skip  (layer 3 + residual)
// =====================================================================
__global__ void gat_finalize_elu(const float* __restrict__ acc,
                                 const float* __restrict__ b,
                                 float* __restrict__ hout, int N) {
  const int t = blockIdx.x * blockDim.x + threadIdx.x;
  if (t >= N * 64) return;
  const float v = acc[t] + b[t & 63];
  hout[t] = v > 0.0f ? v : (__expf(v) - 1.0f);
}

__global__ void gat_finalize_res(const float* __restrict__ acc,
                                 const float* __restrict__ b,
                                 const float* __restrict__ x,
                                 float* __restrict__ out, int N) {
  const int t = blockIdx.x * blockDim.x + threadIdx.x;
  if (t >= N * 64) return;
  out[t] = acc[t] + b[t & 63] + x[t];
}

// =====================================================================
extern "C" void kernel_launch(void* const* d_in, const int* in_sizes, int n_in,
                              void* d_out, int out_size, void* d_ws, size_t ws_size,
                              hipStream_t stream) {
  const float* x  = (const float*)d_in[0];
  const int*   ei = (const int*)d_in[1];
  const float* Wm[3]   = {(const float*)d_in[2],  (const float*)d_in[6],  (const float*)d_in[10]};
  const float* asrc[3] = {(const float*)d_in[3],  (const float*)d_in[7],  (const float*)d_in[11]};
  const float* adst[3] = {(const float*)d_in[4],  (const float*)d_in[8],  (const float*)d_in[12]};
  const float* bias[3] = {(const float*)d_in[5],  (const float*)d_in[9],  (const float*)d_in[13]};

  const int N = in_sizes[0] / 64;
  const int E = in_sizes[1] / 2;
  const int Hs[3] = {8, 8, 1};
  const int Cs[3] = {8, 8, 64};

  // workspace carve-up (all 16B-aligned: every chunk is a multiple of 32B)
  float* base  = (float*)d_ws;
  float* xp    = base;                       // N*64
  float* hA    = xp  + (size_t)N * 64;       // N*64
  float* hB    = hA  + (size_t)N * 64;       // N*64
  float* acc   = hB  + (size_t)N * 64;       // N*64
  float* als   = acc + (size_t)N * 64;       // N*8
  float* ald   = als + (size_t)N * 8;        // N*8
  float* denom = ald + (size_t)N * 8;        // N*8
  unsigned* menc = (unsigned*)(denom + (size_t)N * 8);  // N*8

  const float* hin[3] = {x, hA, hB};
  float*       hup[2] = {hA, hB};

  const int TB = 256;
  const int gGemm = (N + 31) / 32;           // 32 rows per block (8 waves)
  const int gNode64 = (N * 64 + TB - 1) / TB;

  for (int L = 0; L < 3; ++L) {
    const int H = Hs[L], C = Cs[L];
    const int gNH   = (N * H + TB - 1) / TB;
    const int gEH   = ((E + N) * H + TB - 1) / TB;
    const int gMsg  = (((E + N) * 16) + TB - 1) / TB;

    gat_gemm_wmma<<<gGemm, TB, 0, stream>>>(hin[L], Wm[L], xp, N);
    gat_logits<<<gNH, TB, 0, stream>>>(xp, asrc[L], adst[L], als, ald, N, H, C);
    gat_init<<<gNode64, TB, 0, stream>>>(menc, denom, acc, N, H);
    gat_edge_max<<<gEH, TB, 0, stream>>>(ei, E, N, H, als, ald, menc);
    gat_edge_sum<<<gEH, TB, 0, stream>>>(ei, E, N, H, als, ald, menc, denom);
    gat_edge_msg<<<gMsg, TB, 0, stream>>>(ei, E, N, H, C, xp, als, ald, menc, denom, acc);

    if (L < 2) {
      gat_finalize_elu<<<gNode64, TB, 0, stream>>>(acc, bias[L], hup[L], N);
    } else {
      gat_finalize_res<<<gNode64, TB, 0, stream>>>(acc, bias[L], x, (float*)d_out, N);
    }
  }
}